// distort_image_61057255080498
// MI455X (gfx1250) — compile-verified
//
#include <hip/hip_runtime.h>
#include <cstdint>
#include <cstddef>

// ---------------------------------------------------------------------------
// RandAugment-style 2-layer image distortion for MI455X (gfx1250, wave32).
// Memory-bound: ~0.8 GB moved/call -> ~35us floor at 23.3 TB/s.
// ---------------------------------------------------------------------------

namespace {
constexpr int   kH      = 512;
constexpr int   kW      = 512;
constexpr int   kLayers = 2;
constexpr float kMag    = 0.5f;                 // MAG = 5/10
constexpr float kFill   = 0.5f;                 // out-of-bounds gray fill
constexpr float kEnh    = 1.0f + 0.9f * kMag;   // 1.45
// tile geometry for the apply kernel
constexpr int   kTW = 32;   // tile width  (threads x)
constexpr int   kTH = 8;    // tile height (threads y)
constexpr int   kLdsW = kTW + 2;                // 34 (halo for 3x3 stencil)
constexpr int   kLdsH = kTH + 2;                // 10
constexpr int   kLdsElems = kLdsW * kLdsH * 3;  // 1020 floats
// mean reduction: 32 chunks per image so the 3MB stream spreads over 32 WGPs
constexpr int   kChunks = 32;
}

// --- CDNA5 async load-to-LDS (ASYNCcnt path); probe-verified present (the
// round-1 compile error came from inside this block, so the builtin exists).
#if defined(__has_builtin)
#  if __has_builtin(__builtin_amdgcn_global_load_async_to_lds_b32)
#    define AUG_HAVE_ASYNC_LDS 1
#  endif
#endif

#ifdef AUG_HAVE_ASYNC_LDS
typedef __attribute__((address_space(1))) int aug_as1_int;
typedef __attribute__((address_space(3))) int aug_as3_int;

__device__ __forceinline__ void aug_wait_async0() {
#  if __has_builtin(__builtin_amdgcn_s_wait_asynccnt)
  __builtin_amdgcn_s_wait_asynccnt(0);
#  else
  asm volatile("s_wait_asynccnt 0" ::: "memory");
#  endif
}
#endif

__device__ __forceinline__ float aug_clamp01(float v) {
  return fminf(fmaxf(v, 0.0f), 1.0f);
}

// Reference gather(): valid-mask with FILL, clamped indices (always-safe load).
__device__ __forceinline__ void aug_gather(const float* __restrict__ img,
                                           float yy, float xx, float o[3]) {
  const bool valid = (xx >= 0.0f) && (xx <= (float)(kW - 1)) &&
                     (yy >= 0.0f) && (yy <= (float)(kH - 1));
  const int xc = (int)fminf(fmaxf(xx, 0.0f), (float)(kW - 1));
  const int yc = (int)fminf(fmaxf(yy, 0.0f), (float)(kH - 1));
  const float* p = img + ((size_t)yc * kW + xc) * 3;
  o[0] = valid ? p[0] : kFill;
  o[1] = valid ? p[1] : kFill;
  o[2] = valid ? p[2] : kFill;
}

// ---------------------------------------------------------------------------
// Mean stage 1: grid (kChunks, B). Each block reduces 1/32 of one image into
// partials[b*kChunks+chunk]. Skips (writes 0) unless this image's op at this
// layer is contrast. Deterministic: fixed tree order, no float atomics.
// ---------------------------------------------------------------------------
__global__ __launch_bounds__(256) void aug_mean_stage1(
    const float* __restrict__ src, const int* __restrict__ op_ids,
    float* __restrict__ partials, int layer) {
  constexpr int N      = kH * kW * 3;
  constexpr int N4     = N / 4;              // 196608 float4
  constexpr int kChunk4 = N4 / kChunks;      // 6144 float4 per chunk
  const int chunk = blockIdx.x;
  const int b     = blockIdx.y;
  const int op = op_ids[b * kLayers + layer];
  if (op != 6) {  // not contrast: deterministic placeholder, skip the read
    if (threadIdx.x == 0) partials[b * kChunks + chunk] = 0.0f;
    return;
  }
  const float4* img4 = reinterpret_cast<const float4*>(src + (size_t)b * N) +
                       (size_t)chunk * kChunk4;
  float acc = 0.0f;
  for (int i = threadIdx.x; i < kChunk4; i += 256) {
    __builtin_prefetch(img4 + i + 256 * 4, 0, 0);  // global_prefetch_b8
    float4 v = img4[i];
    acc += (v.x + v.y) + (v.z + v.w);
  }
  __shared__ float red[256];
  red[threadIdx.x] = acc;
  __syncthreads();
  for (int off = 128; off > 0; off >>= 1) {
    if (threadIdx.x < (unsigned)off) red[threadIdx.x] += red[threadIdx.x + off];
    __syncthreads();
  }
  if (threadIdx.x == 0) partials[b * kChunks + chunk] = red[0];
}

// ---------------------------------------------------------------------------
// Mean stage 2: one wave32 per image; fixed-order butterfly over 32 partials.
// ---------------------------------------------------------------------------
__global__ __launch_bounds__(32) void aug_mean_stage2(
    const float* __restrict__ partials, float* __restrict__ means) {
  constexpr int N = kH * kW * 3;
  const int b = blockIdx.x;
  float acc = partials[b * kChunks + threadIdx.x];
#pragma unroll
  for (int m = 16; m > 0; m >>= 1) acc += __shfl_xor(acc, m, 32);
  if (threadIdx.x == 0) means[b] = acc / (float)N;
}

// ---------------------------------------------------------------------------
// Apply one augmentation layer. Block = 32x8 pixels of one image
// (blockIdx.z = image). The op is block-uniform -> scalar branching.
// ---------------------------------------------------------------------------
__global__ __launch_bounds__(256) void aug_apply_kernel(
    const float* __restrict__ src, float* __restrict__ dst,
    const int* __restrict__ op_ids, const int* __restrict__ signs,
    const float* __restrict__ means, int layer, int is_final) {
  const int b  = blockIdx.z;
  const int op = op_ids[b * kLayers + layer];
  const int sg = signs[b * kLayers + layer];
  const float s = (float)(2 * sg - 1);

  const float* img = src + (size_t)b * kH * kW * 3;
  float*       out = dst + (size_t)b * kH * kW * 3;

  const int x = blockIdx.x * kTW + threadIdx.x;
  const int y = blockIdx.y * kTH + threadIdx.y;
  const size_t pidx = ((size_t)y * kW + x) * 3;

  __shared__ float tile[kLdsElems];

  float r0, r1, r2;

  if (op < 5) {
    // ---- inverse affine + bilinear resample -------------------------------
    float A = 1.0f, Bc = 0.0f, tx = 0.0f, Cc = 0.0f, D = 1.0f, ty = 0.0f;
    if (op == 0) {              // rotate: th = s * 15 deg
      const float th = s * (30.0f * kMag) * 0.017453292519943295f;
      const float co = cosf(th), si = sinf(th);
      A = co; Bc = si; Cc = -si; D = co;
    } else if (op == 1) {       // shear_x
      Bc = -(s * 0.3f * kMag);
    } else if (op == 2) {       // shear_y
      Cc = -(s * 0.3f * kMag);
    } else if (op == 3) {       // trans_x
      tx = -(s * 0.3f * kMag * (float)kW);
    } else {                    // trans_y
      ty = -(s * 0.3f * kMag * (float)kH);
    }
    const float cx = (kW - 1) * 0.5f, cy = (kH - 1) * 0.5f;
    const float xr = (float)x - cx, yr = (float)y - cy;
    const float xi = A * xr + Bc * yr + tx + cx;
    const float yi = Cc * xr + D * yr + ty + cy;
    const float x0 = floorf(xi), y0 = floorf(yi);
    const float wx = xi - x0, wy = yi - y0;
    const float w00 = (1.0f - wx) * (1.0f - wy);
    const float w10 = wx * (1.0f - wy);
    const float w01 = (1.0f - wx) * wy;
    const float w11 = wx * wy;
    float g00[3], g10[3], g01[3], g11[3];
    aug_gather(img, y0,        x0,        g00);
    aug_gather(img, y0,        x0 + 1.0f, g10);
    aug_gather(img, y0 + 1.0f, x0,        g01);
    aug_gather(img, y0 + 1.0f, x0 + 1.0f, g11);
    r0 = g00[0] * w00 + g10[0] * w10 + g01[0] * w01 + g11[0] * w11;
    r1 = g00[1] * w00 + g10[1] * w10 + g01[1] * w01 + g11[1] * w11;
    r2 = g00[2] * w00 + g10[2] * w10 + g01[2] * w01 + g11[2] * w11;
  } else if (op == 5) {
    // ---- brightness -------------------------------------------------------
    const float f = sg ? kEnh : 1.0f / kEnh;
    const float* p = img + pidx;
    r0 = aug_clamp01(f * p[0]);
    r1 = aug_clamp01(f * p[1]);
    r2 = aug_clamp01(f * p[2]);
  } else if (op == 6) {
    // ---- contrast (uses precomputed per-image mean) -----------------------
    const float f = sg ? kEnh : 1.0f / kEnh;
    const float m = means[b];
    const float* p = img + pidx;
    r0 = aug_clamp01(m + f * (p[0] - m));
    r1 = aug_clamp01(m + f * (p[1] - m));
    r2 = aug_clamp01(m + f * (p[2] - m));
  } else {
    // ---- sharpness: LDS-staged 3x3 edge-clamped smooth --------------------
    const int bx0 = blockIdx.x * kTW, by0 = blockIdx.y * kTH;
    const int tid = threadIdx.y * kTW + threadIdx.x;
    for (int i = tid; i < kLdsElems; i += kTW * kTH) {
      const int pix = i / 3, ch = i - pix * 3;
      const int rr = pix / kLdsW, cc = pix - rr * kLdsW;
      const int gy = min(max(by0 + rr - 1, 0), kH - 1);
      const int gx = min(max(bx0 + cc - 1, 0), kW - 1);
      const float* gp = img + ((size_t)gy * kW + gx) * 3 + ch;
#ifdef AUG_HAVE_ASYNC_LDS
      // AS1 pointer == generic 64-bit address; AS3 pointer == low 32 bits of
      // the generic LDS address (aperture in the high half, per ISA 10.2).
      aug_as1_int* gsrc = (aug_as1_int*)(uintptr_t)gp;
      aug_as3_int* ldst = (aug_as3_int*)(uint32_t)(uintptr_t)&tile[i];
      __builtin_amdgcn_global_load_async_to_lds_b32(gsrc, ldst,
                                                    /*offset=*/0, /*cpol=*/0);
#else
      tile[i] = *gp;
#endif
    }
#ifdef AUG_HAVE_ASYNC_LDS
    aug_wait_async0();
#endif
    __syncthreads();  // op is block-uniform, so all threads reach this

    const float f  = sg ? kEnh : 1.0f / kEnh;
    const int lr = threadIdx.y + 1, lc = threadIdx.x + 1;
    const float wS = 1.0f / 13.0f, wC = 5.0f / 13.0f;
    float res[3];
#pragma unroll
    for (int c = 0; c < 3; ++c) {
      const float sm =
          wS * tile[((lr - 1) * kLdsW + (lc - 1)) * 3 + c] +
          wS * tile[((lr - 1) * kLdsW + lc)       * 3 + c] +
          wS * tile[((lr - 1) * kLdsW + (lc + 1)) * 3 + c] +
          wS * tile[(lr * kLdsW + (lc - 1)) * 3 + c] +
          wC * tile[(lr * kLdsW + lc)       * 3 + c] +
          wS * tile[(lr * kLdsW + (lc + 1)) * 3 + c] +
          wS * tile[((lr + 1) * kLdsW + (lc - 1)) * 3 + c] +
          wS * tile[((lr + 1) * kLdsW + lc)       * 3 + c] +
          wS * tile[((lr + 1) * kLdsW + (lc + 1)) * 3 + c];
      const float v = tile[(lr * kLdsW + lc) * 3 + c];
      res[c] = aug_clamp01(sm + f * (v - sm));
    }
    r0 = res[0]; r1 = res[1]; r2 = res[2];
  }

  float* po = out + pidx;
  if (is_final) {  // final layer output never re-read: NT stores, spare L2
    __builtin_nontemporal_store(r0, po + 0);
    __builtin_nontemporal_store(r1, po + 1);
    __builtin_nontemporal_store(r2, po + 2);
  } else {         // intermediate stays RT so layer-1 pass can hit 192MB L2
    po[0] = r0; po[1] = r1; po[2] = r2;
  }
}

// ---------------------------------------------------------------------------
// Launch: layer0 images->ws, layer1 ws->out.
// ws layout: [means: 1024 f][partials: 3072 f][tmp image: B*H*W*3 f]
// ---------------------------------------------------------------------------
extern "C" void kernel_launch(void* const* d_in, const int* in_sizes, int n_in,
                              void* d_out, int out_size, void* d_ws, size_t ws_size,
                              hipStream_t stream) {
  (void)n_in; (void)out_size; (void)ws_size;
  const float* images = (const float*)d_in[0];
  const int*   op_ids = (const int*)d_in[1];
  const int*   signs  = (const int*)d_in[2];
  float*       out    = (float*)d_out;

  const int B = in_sizes[0] / (kH * kW * 3);

  float* means    = (float*)d_ws;        // B floats
  float* partials = means + 1024;        // B*kChunks floats
  float* tmp      = means + 4096;        // intermediate image (16KB-aligned)

  const dim3 blk(kTW, kTH, 1);
  const dim3 grd(kW / kTW, kH / kTH, B);
  const dim3 mgrd(kChunks, B, 1);

  // layer 0: images -> tmp
  aug_mean_stage1<<<mgrd, 256, 0, stream>>>(images, op_ids, partials, 0);
  aug_mean_stage2<<<B, 32, 0, stream>>>(partials, means);
  aug_apply_kernel<<<grd, blk, 0, stream>>>(images, tmp, op_ids, signs, means,
                                            0, /*is_final=*/0);
  // layer 1: tmp -> out
  aug_mean_stage1<<<mgrd, 256, 0, stream>>>(tmp, op_ids, partials, 1);
  aug_mean_stage2<<<B, 32, 0, stream>>>(partials, means);
  aug_apply_kernel<<<grd, blk, 0, stream>>>(tmp, out, op_ids, signs, means,
                                            1, /*is_final=*/1);
}